// StatDenoiser_20280835571942
// MI455X (gfx1250) — compile-verified
//
#include <hip/hip_runtime.h>

// ---------------------------------------------------------------------------
// Statistical denoiser (11x11 bilateral + Welch membership), MI455X / gfx1250.
//
// Strategy: one 16x16 output tile per 256-thread block. The 26x26 halo of all
// 17 input channels (8 guidance + 3 estimand + 3 variance + 3 image) is staged
// into LDS with FOUR Tensor-Data-Mover descriptors (3-D tile: x=cols, y=rows,
// z=channels; tensor_dim0_stride=512, tensor_dim1_stride=512*512), issued by
// wave 0 and synchronized with s_wait_tensorcnt + workgroup barrier.
// Compute is VALU-bound: per neighbor, 8-ch bilateral distance (fma chain),
// one v_exp_f32, and a division-free membership test
//      t < gamma  <=>  num < T*den,  T = 1 - 0.5/(gamma^2+1)
// Out-of-image neighbors have zero final weight in the reference (zero-padded
// v_j=0 forces w=1 => t=inf => membership 0), so they are simply skipped and
// only the clipped halo rectangle is DMA'd.
// ---------------------------------------------------------------------------

#define IMG_H   512
#define IMG_W   512
#define RAD     5
#define TILE    16
#define HALO    (TILE + 2 * RAD)        // 26
#define NCH     17                      // 8 guidance + 3 est + 3 var + 3 img

typedef unsigned int v4u __attribute__((ext_vector_type(4)));
typedef int          v8i __attribute__((ext_vector_type(8)));
typedef int          v4i __attribute__((ext_vector_type(4)));

typedef __attribute__((address_space(3))) void lds_void_t;

// LDS byte offset of a shared-memory pointer (generic -> AS3, then ptrtoint).
__device__ __forceinline__ unsigned lds_byte_addr(void* p) {
  return (unsigned)(unsigned long long)(lds_void_t*)p;
}

#if __has_builtin(__builtin_amdgcn_exp2f)
#define EXP2F(x) __builtin_amdgcn_exp2f(x)
#else
#define EXP2F(x) exp2f(x)
#endif

// Issue one TENSOR_LOAD_TO_LDS: 3-D tile (w x h x ch) from a CHW float tensor
// with row stride IMG_W and channel stride IMG_W*IMG_H, packed densely in LDS.
__device__ __forceinline__ void tdm_load_tile(const float* gptr, unsigned ldsb,
                                              int tw, int th, int ch) {
  unsigned long long ga = (unsigned long long)(const void*)gptr;
  v4u g0;
  g0.x = 1u;                                            // count=1, no gather
  g0.y = ldsb;                                          // lds_addr (bytes)
  g0.z = (unsigned)ga;                                  // global_addr[31:0]
  g0.w = (unsigned)((ga >> 32) & 0x01FFFFFFull)         // global_addr[56:32]
       | (2u << 30);                                    // type=2 ("image")
  v8i g1;
  g1[0] = (int)(2u << 16);                              // data_size=4B, mask=0
  g1[1] = (int)(((unsigned)tw) << 16);                  // tensor_dim0[15:0]
  g1[2] = (int)(((unsigned)th) << 16);                  // td0 hi=0 | td1[15:0]
  g1[3] = (int)(((unsigned)tw) << 16);                  // td1 hi=0 | tile_dim0
  g1[4] = (int)(((unsigned)th) | (((unsigned)ch) << 16)); // tile_dim1|tile_dim2
  g1[5] = IMG_W;                                        // dim0 stride lo
  g1[6] = 0;                                            // dim0 hi | dim1 s[15:0]
  g1[7] = (int)((unsigned)(IMG_W * IMG_H) >> 16);       // dim1 stride[47:16]
  v4i g2; g2[0] = ch; g2[1] = 0; g2[2] = 0; g2[3] = 0;  // tensor_dim2=ch
  v4i g3; g3[0] = 0;  g3[1] = 0; g3[2] = 0; g3[3] = 0;
#if __clang_major__ >= 23
  v8i g4; g4[0]=0; g4[1]=0; g4[2]=0; g4[3]=0; g4[4]=0; g4[5]=0; g4[6]=0; g4[7]=0;
  __builtin_amdgcn_tensor_load_to_lds(g0, g1, g2, g3, g4, 0);
#else
  __builtin_amdgcn_tensor_load_to_lds(g0, g1, g2, g3, 0);
#endif
}

__global__ __launch_bounds__(256)
void StatDenoiser_gfx1250_kernel(const float* __restrict__ image,
                                 const float* __restrict__ guidance,
                                 const float* __restrict__ est,
                                 const float* __restrict__ var,
                                 const int*   __restrict__ spp_p,
                                 float*       __restrict__ out) {
  __shared__ float smem[NCH * HALO * HALO];             // 45.97 KB

  const int x0  = blockIdx.x * TILE;
  const int y0  = blockIdx.y * TILE;
  const int gx0 = x0 - RAD, gy0 = y0 - RAD;             // halo origin (may be <0)
  const int cx0 = gx0 < 0 ? 0 : gx0;                    // clipped halo rectangle
  const int cy0 = gy0 < 0 ? 0 : gy0;
  int cx1 = gx0 + HALO; if (cx1 > IMG_W) cx1 = IMG_W;
  int cy1 = gy0 + HALO; if (cy1 > IMG_H) cy1 = IMG_H;
  const int tw = cx1 - cx0;
  const int th = cy1 - cy0;
  const int hw = tw * th;

  // ---- Stage halo into LDS via TDM (wave 0 only; TENSORcnt is per-wave) ----
  if (threadIdx.x < 32) {
    const unsigned base = lds_byte_addr((void*)smem);
    const int goff = cy0 * IMG_W + cx0;
    tdm_load_tile(guidance + goff, base,                        tw, th, 8);
    tdm_load_tile(est      + goff, base + 4u * (unsigned)( 8*hw), tw, th, 3);
    tdm_load_tile(var      + goff, base + 4u * (unsigned)(11*hw), tw, th, 3);
    tdm_load_tile(image    + goff, base + 4u * (unsigned)(14*hw), tw, th, 3);
    __builtin_amdgcn_s_wait_tensorcnt(0);
  }
  __syncthreads();

  // ---- Welch threshold from runtime spp (z = ndtri(1 - alpha/2), alpha=.005)
  const float z  = 2.80703378f;
  const float z2 = z * z, z3 = z2 * z, z5 = z3 * z2;
  const float df = (float)(spp_p[0] + spp_p[0] - 2);
  const float gamma = z + ((z3 + z) * 0.25f) / df
                        + ((5.f * z5 + 16.f * z3 + 3.f * z) * (1.f / 96.f)) / (df * df);
  const float G2 = gamma * gamma;
  const float T  = 1.f - 0.5f / (G2 + 1.f);             // num < T*den  <=>  t < gamma

  // ---- Per-thread pixel ----
  const int tx = threadIdx.x & (TILE - 1);
  const int ty = threadIdx.x >> 4;
  const int x = x0 + tx, y = y0 + ty;

  const float* sg = smem;                               // [8][th][tw]
  const float* se = smem +  8 * hw;                     // [3][th][tw]
  const float* sv = smem + 11 * hw;                     // [3][th][tw]
  const float* si = smem + 14 * hw;                     // [3][th][tw]
  const int cidx = (y - cy0) * tw + (x - cx0);

  float gc[8];
#pragma unroll
  for (int c = 0; c < 8; ++c) gc[c] = sg[c * hw + cidx];
  float ec[3], vc[3];
#pragma unroll
  for (int c = 0; c < 3; ++c) { ec[c] = se[c * hw + cidx]; vc[c] = sv[c * hw + cidx]; }

  const float sig[8] = {0.1f, 0.1f, 50.f, 50.f, 50.f, 10.f, 10.f, 10.f};

  float acc0 = 0.f, acc1 = 0.f, acc2 = 0.f, sumw = 0.f;
  for (int dy = -RAD; dy <= RAD; ++dy) {
    const int ny = y + dy;
    if ((unsigned)ny >= IMG_H) continue;                // zero-padded => weight 0
    const int rowb = (ny - cy0) * tw;
    for (int dx = -RAD; dx <= RAD; ++dx) {
      const int nx = x + dx;
      if ((unsigned)nx >= IMG_W) continue;
      const int idx = rowb + (nx - cx0);

      // Bilateral guidance distance (8 channels).
      float r = 0.f;
#pragma unroll
      for (int c = 0; c < 8; ++c) {
        const float d = sg[c * hw + idx] - gc[c];
        r = fmaf(d * d, sig[c], r);
      }
      const float bw = EXP2F(r * -0.72134752f);         // exp(-0.5*r)

      // Membership: all 3 channels must pass t < gamma (division-free form).
      bool mem = true;
#pragma unroll
      for (int c = 0; c < 3; ++c) {
        const float ej  = se[c * hw + idx];
        const float vj  = sv[c * hw + idx];
        const float d   = ec[c] - ej;
        const float d2  = d * d;
        const float s   = vc[c] + vj;
        const float num = 2.f * d2 + s;
        const float den = num + s;                      // 2*d2 + 2*s
        bool pass = (num < T * den) || (den == 0.f);    // den==0 => res=0.5 => pass
        if (((vc[c] == 0.f) || (vj == 0.f)) && (d != 0.f)) pass = false; // w=1 => t=inf
        mem = mem && pass;
      }

      const float wgt = mem ? bw : 0.f;
      sumw += wgt;
      acc0 = fmaf(wgt, si[idx],          acc0);
      acc1 = fmaf(wgt, si[hw + idx],     acc1);
      acc2 = fmaf(wgt, si[2 * hw + idx], acc2);
    }
  }

  const float inv = 1.f / fmaxf(sumw, 1e-10f);          // center weight is 1, sumw>=1
  const int o = y * IMG_W + x;
  out[o]                   = acc0 * inv;
  out[IMG_H * IMG_W + o]   = acc1 * inv;
  out[2 * IMG_H * IMG_W + o] = acc2 * inv;
}

extern "C" void kernel_launch(void* const* d_in, const int* in_sizes, int n_in,
                              void* d_out, int out_size, void* d_ws, size_t ws_size,
                              hipStream_t stream) {
  (void)in_sizes; (void)n_in; (void)out_size; (void)d_ws; (void)ws_size;
  const float* image    = (const float*)d_in[0];   // (1,3,512,512)
  const float* guidance = (const float*)d_in[1];   // (1,8,512,512)
  const float* est      = (const float*)d_in[2];   // (1,3,512,512)
  const float* var      = (const float*)d_in[3];   // (1,3,512,512)
  const int*   spp      = (const int*)d_in[4];     // scalar
  float* out = (float*)d_out;                      // (1,3,512,512)

  dim3 grid(IMG_W / TILE, IMG_H / TILE);           // 32 x 32 tiles
  StatDenoiser_gfx1250_kernel<<<grid, 256, 0, stream>>>(image, guidance, est,
                                                        var, spp, out);
}